// SURGE_34110630265583
// MI455X (gfx1250) — compile-verified
//
#include <hip/hip_runtime.h>
#include <math.h>

typedef __attribute__((ext_vector_type(2))) float v2f;
typedef __attribute__((ext_vector_type(8))) float v8f;

#define BB 128
#define LL 256
#define EE 64
#define HH 64
#define NEGV (-4294967295.0f)
#define EPS7 1e-7f

// ---------------------------------------------------------------------------
// Stage 1: embedding gather + masked X, metric-normalized Xf
// ---------------------------------------------------------------------------
__global__ void k_embed(const int* __restrict__ mid_his, const float* __restrict__ mask,
                        const float* __restrict__ emb, const float* __restrict__ w_metric,
                        float* __restrict__ X, float* __restrict__ Xf) {
  int row = blockIdx.x;            // b*L + l
  int t = threadIdx.x;             // 0..63
  __shared__ float red[64];
  int idx = mid_his[row];
  float m = mask[row];
  float x = emb[(size_t)idx * EE + t] * m;
  X[(size_t)row * EE + t] = x;
  float xw = x * w_metric[t];
  red[t] = xw * xw;
  __syncthreads();
  for (int s = 32; s > 0; s >>= 1) { if (t < s) red[t] += red[t + s]; __syncthreads(); }
  float rn = rsqrtf(fmaxf(red[0], 1e-12f));
  Xf[(size_t)row * EE + t] = xw * rn;
}

__global__ void k_item(const int* __restrict__ mid_batch, const float* __restrict__ emb,
                       float* __restrict__ item_eb) {
  int b = blockIdx.x; int t = threadIdx.x;
  item_eb[b * EE + t] = emb[(size_t)mid_batch[b] * EE + t];
}

// ---------------------------------------------------------------------------
// WMMA fp32 similarity: S[b] = Xf[b] * Xf[b]^T  (per-batch 256x256, K=64)
// one wave per 16x16 tile; V_WMMA_F32_16X16X4_F32
// ---------------------------------------------------------------------------
__global__ void k_sim(const float* __restrict__ Xf, float* __restrict__ S) {
  int b = blockIdx.x, tm = blockIdx.y, tn = blockIdx.z;
  int lane = threadIdx.x;
  int r = lane & 15, hi = lane >> 4;
  const float* Xb = Xf + (size_t)b * LL * EE;
  const float* Arow = Xb + (size_t)(tm * 16 + r) * EE;
  const float* Brow = Xb + (size_t)(tn * 16 + r) * EE;   // B = Xf^T -> B[k][n] = Xf[n][k]
  v8f acc = {};
  for (int kk = 0; kk < EE; kk += 4) {
    v2f a, bb;
    a.x = Arow[kk + 2 * hi + 0]; a.y = Arow[kk + 2 * hi + 1];
    bb.x = Brow[kk + 2 * hi + 0]; bb.y = Brow[kk + 2 * hi + 1];
    acc = __builtin_amdgcn_wmma_f32_16x16x4_f32(false, a, false, bb, (short)0, acc, false, false);
  }
  float* Sb = S + (size_t)b * LL * LL;
#pragma unroll
  for (int i = 0; i < 8; ++i)
    Sb[(size_t)(tm * 16 + hi * 8 + i) * LL + tn * 16 + r] = acc[i];
}

// ---------------------------------------------------------------------------
// Row min/max normalize, mask, edge count
// ---------------------------------------------------------------------------
__global__ void k_rownorm(float* __restrict__ S, const float* __restrict__ mask,
                          int* __restrict__ edges) {
  int row = blockIdx.x;            // b*L + l
  int b = row >> 8;
  int t = threadIdx.x;             // m
  __shared__ float rmin[256], rmax[256];
  __shared__ int cnt[256];
  float v = S[(size_t)row * LL + t];
  rmin[t] = v; rmax[t] = v;
  __syncthreads();
  for (int s = 128; s > 0; s >>= 1) {
    if (t < s) { rmin[t] = fminf(rmin[t], rmin[t + s]); rmax[t] = fmaxf(rmax[t], rmax[t + s]); }
    __syncthreads();
  }
  float mn = rmin[0], mx = rmax[0];
  float den = mx - mn;
  float nv = (den > 0.f) ? (v - mn) / den : 0.f;   // NaN (0/0) -> 0
  nv *= mask[row] * mask[b * LL + t];
  S[(size_t)row * LL + t] = nv;
  cnt[t] = (nv != 0.f) ? 1 : 0;
  __syncthreads();
  for (int s = 128; s > 0; s >>= 1) { if (t < s) cnt[t] += cnt[t + s]; __syncthreads(); }
  if (t == 0) atomicAdd(&edges[b], cnt[0]);
}

// ---------------------------------------------------------------------------
// Exact rank selection: thr[b] = (keep)-th largest of S[b] (0-indexed),
// keep = ceil(num_edges/2). Radix-select on uint bit patterns (values >= 0).
// ---------------------------------------------------------------------------
__global__ void k_select(const float* __restrict__ S, const int* __restrict__ edges,
                         float* __restrict__ thr) {
  int b = blockIdx.x; int t = threadIdx.x;
  __shared__ unsigned hist[256];
  __shared__ unsigned sh_prefix;
  __shared__ int sh_rank;
  const float* Sb = S + (size_t)b * LL * LL;
  if (t == 0) { sh_rank = (edges[b] + 1) >> 1; sh_prefix = 0u; }
  __syncthreads();
  for (int byt = 3; byt >= 0; --byt) {
    hist[t] = 0u;
    __syncthreads();
    unsigned himask = (byt == 3) ? 0u : (0xFFFFFFFFu << (8 * (byt + 1)));
    unsigned prefix = sh_prefix;
    for (int i = t; i < LL * LL; i += 256) {
      unsigned u = __float_as_uint(Sb[i]);
      if ((u & himask) == prefix)
        atomicAdd(&hist[(u >> (8 * byt)) & 255u], 1u);
    }
    __syncthreads();
    if (t == 0) {
      int k = sh_rank, cum = 0, bin = 0;
      for (int j = 255; j >= 0; --j) {
        int c = (int)hist[j];
        if (cum + c > k) { bin = j; k -= cum; break; }
        cum += c;
      }
      sh_rank = k;
      sh_prefix = prefix | ((unsigned)bin << (8 * byt));
    }
    __syncthreads();
  }
  if (t == 0) thr[b] = __uint_as_float(sh_prefix);
}

// ---------------------------------------------------------------------------
// A_bool (diag forced 1) + degree d = sqrt(rowsum)+EPS ; An = A_bool/(d_l*d_m)
// ---------------------------------------------------------------------------
__global__ void k_abool(const float* __restrict__ S, const float* __restrict__ thr,
                        float* __restrict__ Abool, float* __restrict__ ds) {
  int row = blockIdx.x; int b = row >> 8; int l = row & 255;
  int t = threadIdx.x;
  __shared__ float red[256];
  float th = thr[b];
  float s = S[(size_t)row * LL + t];
  float a = (t == l) ? 1.f : ((s > th) ? 1.f : 0.f);
  Abool[(size_t)row * LL + t] = a;
  red[t] = a;
  __syncthreads();
  for (int st = 128; st > 0; st >>= 1) { if (t < st) red[t] += red[t + st]; __syncthreads(); }
  if (t == 0) ds[row] = sqrtf(red[0]) + EPS7;
}

__global__ void k_an(const float* __restrict__ Abool, const float* __restrict__ ds,
                     float* __restrict__ An) {
  int row = blockIdx.x; int b = row >> 8;
  int t = threadIdx.x;
  An[(size_t)row * LL + t] = Abool[(size_t)row * LL + t] / (ds[row] * ds[b * LL + t]);
}

// ---------------------------------------------------------------------------
// WMMA batched: C[b] = A[b](LxL) @ Bm[b](LxN), N multiple of 16
// ---------------------------------------------------------------------------
__global__ void k_mm_batched(const float* __restrict__ A, const float* __restrict__ Bm,
                             float* __restrict__ C, int Ncols) {
  int b = blockIdx.x, tm = blockIdx.y, tn = blockIdx.z;
  int lane = threadIdx.x; int r = lane & 15, hi = lane >> 4;
  const float* Ab = A + (size_t)b * LL * LL + (size_t)(tm * 16 + r) * LL;
  const float* Bb = Bm + (size_t)b * LL * Ncols + tn * 16 + r;
  v8f acc = {};
  for (int kk = 0; kk < LL; kk += 4) {
    v2f a, bb;
    a.x = Ab[kk + 2 * hi]; a.y = Ab[kk + 2 * hi + 1];
    bb.x = Bb[(size_t)(kk + 2 * hi) * Ncols]; bb.y = Bb[(size_t)(kk + 2 * hi + 1) * Ncols];
    acc = __builtin_amdgcn_wmma_f32_16x16x4_f32(false, a, false, bb, (short)0, acc, false, false);
  }
  float* Cb = C + (size_t)b * LL * Ncols;
#pragma unroll
  for (int i = 0; i < 8; ++i)
    Cb[(size_t)(tm * 16 + hi * 8 + i) * Ncols + tn * 16 + r] = acc[i];
}

// ---------------------------------------------------------------------------
// WMMA row-major generic: C(MxN) = A(MxK) @ W(KxN), fused epilogues
// mode 0: plain   mode 1: relu(.+b)*gamma/sqrt(1.0001)+beta   mode 2: lrelu(.+b+add)
// ---------------------------------------------------------------------------
__global__ void k_mm_rows(const float* __restrict__ A, const float* __restrict__ W,
                          float* __restrict__ C, int K, int N, int mode,
                          const float* __restrict__ bias, const float* __restrict__ gamma,
                          const float* __restrict__ beta, const float* __restrict__ addsrc) {
  int tm = blockIdx.x, tn = blockIdx.y;
  int lane = threadIdx.x; int r = lane & 15, hi = lane >> 4;
  const float* Ar = A + (size_t)(tm * 16 + r) * K;
  const float* Wc = W + tn * 16 + r;
  v8f acc = {};
  for (int kk = 0; kk < K; kk += 4) {
    v2f a, bb;
    a.x = Ar[kk + 2 * hi]; a.y = Ar[kk + 2 * hi + 1];
    bb.x = Wc[(size_t)(kk + 2 * hi) * N]; bb.y = Wc[(size_t)(kk + 2 * hi + 1) * N];
    acc = __builtin_amdgcn_wmma_f32_16x16x4_f32(false, a, false, bb, (short)0, acc, false, false);
  }
  int col = tn * 16 + r;
#pragma unroll
  for (int i = 0; i < 8; ++i) {
    int row = tm * 16 + hi * 8 + i;
    float v = acc[i];
    if (mode == 1) {
      v = fmaxf(v + bias[col], 0.f);
      v = v * (gamma[col] * rsqrtf(1.0001f)) + beta[col];
    } else if (mode == 2) {
      v = v + bias[col] + addsrc[(size_t)row * N + col];
      v = (v >= 0.f) ? v : 0.2f * v;
    }
    C[(size_t)row * N + col] = v;
  }
}

// ---------------------------------------------------------------------------
// Attention helpers: concat [ai, q, ai-q, ai*q]; per-row score; masked softmax
// ---------------------------------------------------------------------------
__global__ void k_hcat(const float* __restrict__ ai, const float* __restrict__ Q, int q3d,
                       float* __restrict__ hcat) {
  int row = blockIdx.x;            // b*L + l
  int t = threadIdx.x;             // 0..63
  float a = ai[(size_t)row * EE + t];
  float q = q3d ? Q[(size_t)row * EE + t] : Q[(size_t)(row >> 8) * EE + t];
  float* h = hcat + (size_t)row * (4 * EE);
  h[t] = a; h[EE + t] = q; h[2 * EE + t] = a - q; h[3 * EE + t] = a * q;
}

__global__ void k_score(const float* __restrict__ h, const float* __restrict__ W2,
                        const float* __restrict__ b2, float* __restrict__ sraw, int M) {
  int row = blockIdx.x * blockDim.x + threadIdx.x;
  if (row >= M) return;
  float acc = b2[0];
  for (int j = 0; j < HH; ++j) acc += h[(size_t)row * HH + j] * W2[j];
  sraw[row] = fmaxf(acc, 0.f);
}

__global__ void k_softmax(const float* __restrict__ sraw, const float* __restrict__ mask,
                          float* __restrict__ f) {
  int b = blockIdx.x; int t = threadIdx.x;
  __shared__ float red[256];
  float m = mask[b * LL + t];
  float v = (m > 0.f) ? sraw[b * LL + t] : NEGV;
  red[t] = v; __syncthreads();
  for (int s = 128; s > 0; s >>= 1) { if (t < s) red[t] = fmaxf(red[t], red[t + s]); __syncthreads(); }
  float mx = red[0]; __syncthreads();
  float e = expf(v - mx);
  red[t] = e; __syncthreads();
  for (int s = 128; s > 0; s >>= 1) { if (t < s) red[t] += red[t + s]; __syncthreads(); }
  f[b * LL + t] = e / red[0];
}

// ---------------------------------------------------------------------------
// Edge-weight softmax: Egr = softmax_m( A_bool==1 ? lrelu(A_bool*(f1[m]+f2[l])) : NEG )
// ---------------------------------------------------------------------------
__global__ void k_egr(const float* __restrict__ Abool, const float* __restrict__ f1,
                      const float* __restrict__ f2, float* __restrict__ Egr) {
  int row = blockIdx.x; int b = row >> 8;
  int t = threadIdx.x;
  __shared__ float red[256];
  float a = Abool[(size_t)row * LL + t];
  float e = a * (f1[b * LL + t] + f2[row]);
  e = (e >= 0.f) ? e : 0.2f * e;
  float v = (a == 1.f) ? e : NEGV;
  red[t] = v; __syncthreads();
  for (int s = 128; s > 0; s >>= 1) { if (t < s) red[t] = fmaxf(red[t], red[t + s]); __syncthreads(); }
  float mx = red[0]; __syncthreads();
  float ex = expf(v - mx);
  red[t] = ex; __syncthreads();
  for (int s = 128; s > 0; s >>= 1) { if (t < s) red[t] += red[t + s]; __syncthreads(); }
  Egr[(size_t)row * LL + t] = ex / red[0];
}

// ---------------------------------------------------------------------------
// Cluster score softmax, bitonic top-k threshold, pooled mask + stable compaction
// ---------------------------------------------------------------------------
__global__ void k_cluster(const float* __restrict__ g1, const float* __restrict__ g2,
                          const float* __restrict__ mask, float* __restrict__ cs,
                          float* __restrict__ wpool, int* __restrict__ kept_idx,
                          float* __restrict__ kept_a, int* __restrict__ nkeep) {
  int b = blockIdx.x; int t = threadIdx.x;
  __shared__ float red[256];
  __shared__ float sc[256];
  __shared__ float srt[256];
  __shared__ float sh_target;
  float m = mask[b * LL + t];
  float v = (m > 0.f) ? (g1[b * LL + t] + g2[b * LL + t]) : NEGV;
  red[t] = v; __syncthreads();
  for (int s = 128; s > 0; s >>= 1) { if (t < s) red[t] = fmaxf(red[t], red[t + s]); __syncthreads(); }
  float mx = red[0]; __syncthreads();
  float e = expf(v - mx);
  red[t] = e; __syncthreads();
  for (int s = 128; s > 0; s >>= 1) { if (t < s) red[t] += red[t + s]; __syncthreads(); }
  float c = (e / red[0]) * m;
  sc[t] = c; srt[t] = c;
  __syncthreads();
  red[t] = m; __syncthreads();
  for (int s = 128; s > 0; s >>= 1) { if (t < s) red[t] += red[t + s]; __syncthreads(); }
  int keep2 = (int)fminf(red[0], 10.0f);
  __syncthreads();
  // ascending bitonic sort of srt
  for (int k = 2; k <= 256; k <<= 1) {
    for (int j = k >> 1; j > 0; j >>= 1) {
      int ixj = t ^ j;
      if (ixj > t) {
        float x0 = srt[t], x1 = srt[ixj];
        bool up = ((t & k) == 0);
        if (up ? (x0 > x1) : (x0 < x1)) { srt[t] = x1; srt[ixj] = x0; }
      }
      __syncthreads();
    }
  }
  if (t == 0) sh_target = srt[255 - keep2] + EPS7;   // descending index keep2
  __syncthreads();
  float target = sh_target;
  float nm = (c > target) ? 1.f : 0.f;
  cs[b * LL + t] = c;
  wpool[b * LL + t] = c * nm;
  red[t] = nm;
  __syncthreads();
  if (t == 0) {
    int cidx = 0;
    for (int l = 0; l < LL; ++l)
      if (red[l] > 0.f) { kept_idx[b * LL + cidx] = l; kept_a[b * LL + cidx] = sc[l]; ++cidx; }
    nkeep[b] = cidx;
  }
}

__global__ void k_readout(const float* __restrict__ Xc, const float* __restrict__ wpool,
                          float* __restrict__ readout) {
  int b = blockIdx.x; int t = threadIdx.x;      // 64
  float acc = 0.f;
  for (int l = 0; l < LL; ++l)
    acc += Xc[((size_t)b * LL + l) * EE + t] * wpool[b * LL + l];
  readout[b * EE + t] = acc;
}

// ---------------------------------------------------------------------------
// AUGRU over kept (compacted) steps only; un-kept steps leave h unchanged.
// ---------------------------------------------------------------------------
__global__ void k_augru(const float* __restrict__ Xc, const int* __restrict__ kept_idx,
                        const float* __restrict__ kept_a, const int* __restrict__ nkeep,
                        const float* __restrict__ Wg, const float* __restrict__ bg,
                        const float* __restrict__ Wc, const float* __restrict__ bc,
                        float* __restrict__ fs) {
  int b = blockIdx.x; int t = threadIdx.x;      // 128
  __shared__ float xs[64], hs[64], ru[128];
  if (t < 64) hs[t] = 0.f;
  __syncthreads();
  int n = nkeep[b];
  for (int step = 0; step < n; ++step) {
    int idx = kept_idx[b * LL + step];
    float alpha = kept_a[b * LL + step];
    if (t < 64) xs[t] = Xc[((size_t)b * LL + idx) * EE + t];
    __syncthreads();
    float acc = bg[t];
    for (int j = 0; j < 64; ++j) acc += xs[j] * Wg[(size_t)j * 128 + t];
    for (int j = 0; j < 64; ++j) acc += hs[j] * Wg[(size_t)(64 + j) * 128 + t];
    ru[t] = 1.f / (1.f + expf(-acc));
    __syncthreads();
    float hnew = 0.f;
    if (t < 64) {
      float cacc = bc[t];
      for (int j = 0; j < 64; ++j) cacc += xs[j] * Wc[(size_t)j * 64 + t];
      for (int j = 0; j < 64; ++j) cacc += (ru[j] * hs[j]) * Wc[(size_t)(64 + j) * 64 + t];
      float cc = tanhf(cacc);
      float u = (1.f - alpha) * ru[64 + t];
      hnew = u * hs[t] + (1.f - u) * cc;
    }
    __syncthreads();
    if (t < 64) hs[t] = hnew;
    __syncthreads();
  }
  if (t < 64) fs[b * HH + t] = hs[t];
}

__global__ void k_out(const float* __restrict__ fs, const float* __restrict__ readout,
                      const float* __restrict__ W_out, const float* __restrict__ b_out,
                      float* __restrict__ out) {
  int b = blockIdx.x; int t = threadIdx.x;      // 64
  float acc = b_out[t];
  for (int j = 0; j < 64; ++j) acc += fs[b * HH + j] * W_out[(size_t)j * HH + t];
  for (int j = 0; j < 64; ++j) acc += readout[b * EE + j] * W_out[(size_t)(64 + j) * HH + t];
  out[b * HH + t] = acc;
}

// ---------------------------------------------------------------------------
// Host orchestration
// ---------------------------------------------------------------------------
extern "C" void kernel_launch(void* const* d_in, const int* in_sizes, int n_in,
                              void* d_out, int out_size, void* d_ws, size_t ws_size,
                              hipStream_t stream) {
  (void)in_sizes; (void)n_in; (void)out_size; (void)ws_size;
  const int*   mid_his   = (const int*)d_in[0];
  const int*   mid_batch = (const int*)d_in[1];
  const float* mask      = (const float*)d_in[2];
  const float* emb       = (const float*)d_in[3];
  const float* w_metric  = (const float*)d_in[4];
  const float* att_mat1  = (const float*)d_in[5];
  const float* W1a       = (const float*)d_in[6];
  const float* b1a       = (const float*)d_in[7];
  const float* g1a       = (const float*)d_in[8];
  const float* be1a      = (const float*)d_in[9];
  const float* W2a       = (const float*)d_in[10];
  const float* b2a       = (const float*)d_in[11];
  const float* att_mat2  = (const float*)d_in[12];
  const float* W1b       = (const float*)d_in[13];
  const float* b1b       = (const float*)d_in[14];
  const float* g1b       = (const float*)d_in[15];
  const float* be1b      = (const float*)d_in[16];
  const float* W2b       = (const float*)d_in[17];
  const float* b2b       = (const float*)d_in[18];
  const float* W_agg     = (const float*)d_in[19];
  const float* b_agg     = (const float*)d_in[20];
  const float* Wg        = (const float*)d_in[21];
  const float* bg        = (const float*)d_in[22];
  const float* Wc        = (const float*)d_in[23];
  const float* bc        = (const float*)d_in[24];
  const float* W_out     = (const float*)d_in[25];
  const float* b_out     = (const float*)d_in[26];
  float* out = (float*)d_out;

  const size_t BL = (size_t)BB * LL;       // 32768 rows
  char* p = (char*)d_ws;
  auto alloc = [&](size_t bytes) { char* r = p; p += (bytes + 255) & ~(size_t)255; return r; };
  float* X       = (float*)alloc(BL * EE * 4);          // 8 MB
  float* Xf      = (float*)alloc(BL * EE * 4);          // 8 MB
  float* S       = (float*)alloc(BL * LL * 4);          // 32 MB (later reused as Egr)
  float* Abool   = (float*)alloc(BL * LL * 4);          // 32 MB
  float* An      = (float*)alloc(BL * LL * 4);          // 32 MB
  float* Xq      = (float*)alloc(BL * EE * 4);          // 8 MB
  float* tmpLE   = (float*)alloc(BL * EE * 4);          // 8 MB
  float* hcat    = (float*)alloc(BL * 4 * EE * 4);      // 32 MB
  float* ai      = (float*)alloc(BL * EE * 4);          // 8 MB
  float* hbuf    = (float*)alloc(BL * HH * 4);          // 8 MB
  float* Xc      = (float*)alloc(BL * EE * 4);          // 8 MB
  float* item_eb = (float*)alloc((size_t)BB * EE * 4);
  float* f1      = (float*)alloc(BL * 4);
  float* f2      = (float*)alloc(BL * 4);
  float* g1      = (float*)alloc(BL * 4);
  float* g2      = (float*)alloc(BL * 4);
  float* sraw    = (float*)alloc(BL * 4);
  float* cs      = (float*)alloc(BL * 4);
  float* wpool   = (float*)alloc(BL * 4);
  float* dsq     = (float*)alloc(BL * 4);
  float* thr     = (float*)alloc((size_t)BB * 4);
  int*   edges   = (int*)  alloc((size_t)BB * 4);
  int*   kept_idx= (int*)  alloc(BL * 4);
  float* kept_a  = (float*)alloc(BL * 4);
  int*   nkeep   = (int*)  alloc((size_t)BB * 4);
  float* readout = (float*)alloc((size_t)BB * EE * 4);
  float* fs      = (float*)alloc((size_t)BB * HH * 4);

  const int M16 = (int)(BL / 16);          // 2048 row-tiles
  dim3 wv(32);

  // 1) gather + metric normalize
  k_embed<<<(int)BL, 64, 0, stream>>>(mid_his, mask, emb, w_metric, X, Xf);
  k_item<<<BB, 64, 0, stream>>>(mid_batch, emb, item_eb);

  // 2) similarity + normalize + exact rank threshold
  hipMemsetAsync(edges, 0, (size_t)BB * 4, stream);
  k_sim<<<dim3(BB, 16, 16), wv, 0, stream>>>(Xf, S);
  k_rownorm<<<(int)BL, 256, 0, stream>>>(S, mask, edges);
  k_select<<<BB, 256, 0, stream>>>(S, edges, thr);
  k_abool<<<(int)BL, 256, 0, stream>>>(S, thr, Abool, dsq);
  k_an<<<(int)BL, 256, 0, stream>>>(Abool, dsq, An);

  // 3) Xq = An @ (An @ X)
  k_mm_batched<<<dim3(BB, 16, 4), wv, 0, stream>>>(An, X, tmpLE, EE);
  k_mm_batched<<<dim3(BB, 16, 4), wv, 0, stream>>>(An, tmpLE, Xq, EE);

  auto run_attn = [&](const float* KV, const float* Q, int q3d, const float* am,
                      const float* W1, const float* b1, const float* ga, const float* be,
                      const float* W2, const float* b2, float* fout) {
    k_mm_rows<<<dim3(M16, 4), wv, 0, stream>>>(KV, am, ai, EE, EE, 0, nullptr, nullptr, nullptr, nullptr);
    k_hcat<<<(int)BL, 64, 0, stream>>>(ai, Q, q3d, hcat);
    k_mm_rows<<<dim3(M16, 4), wv, 0, stream>>>(hcat, W1, hbuf, 4 * EE, HH, 1, b1, ga, be, nullptr);
    k_score<<<(int)(BL / 256), 256, 0, stream>>>(hbuf, W2, b2, sraw, (int)BL);
    k_softmax<<<BB, 256, 0, stream>>>(sraw, mask, fout);
  };

  // 4) fusion attention + edge softmax + aggregation
  run_attn(X, Xq, 1, att_mat1, W1a, b1a, g1a, be1a, W2a, b2a, f1);
  run_attn(X, X,  1, att_mat2, W1b, b1b, g1b, be1b, W2b, b2b, f2);
  k_egr<<<(int)BL, 256, 0, stream>>>(Abool, f1, f2, S);          // S := Egr
  k_mm_batched<<<dim3(BB, 16, 4), wv, 0, stream>>>(S, X, tmpLE, EE);
  k_mm_rows<<<dim3(M16, 4), wv, 0, stream>>>(tmpLE, W_agg, Xc, EE, EE, 2, b_agg, nullptr, nullptr, X);

  // 5) extraction attention: Xq2 = An @ (An @ Xc)
  k_mm_batched<<<dim3(BB, 16, 4), wv, 0, stream>>>(An, Xc, tmpLE, EE);
  k_mm_batched<<<dim3(BB, 16, 4), wv, 0, stream>>>(An, tmpLE, Xq, EE);   // Xq := Xq2
  run_attn(Xc, Xq, 1, att_mat1, W1a, b1a, g1a, be1a, W2a, b2a, g1);
  run_attn(Xc, item_eb, 0, att_mat2, W1b, b1b, g1b, be1b, W2b, b2b, g2);

  // 6) pooling + readout + AUGRU + head
  k_cluster<<<BB, 256, 0, stream>>>(g1, g2, mask, cs, wpool, kept_idx, kept_a, nkeep);
  k_readout<<<BB, 64, 0, stream>>>(Xc, wpool, readout);
  k_augru<<<BB, 128, 0, stream>>>(Xc, kept_idx, kept_a, nkeep, Wg, bg, Wc, bc, fs);
  k_out<<<BB, 64, 0, stream>>>(fs, readout, W_out, b_out, out);
}